// GIN_29978871726578
// MI455X (gfx1250) — compile-verified
//
#include <hip/hip_runtime.h>

// GIN forward for MI455X (gfx1250, wave32).
// Pipeline (all on `stream`, graph-capture safe, no host-side sync):
//   B0 = (1+eps0)*x;  B0 += SpMM(vals, x)            [f32 HW atomics, L2-resident]
//   B1 = relu(B0 @ W1a^T + b1a)    [WMMA f32 16x16x4]
//   B0 = B1 @ W2a^T + b2a          (h1)
//   B1 = (1+eps1)*B0; B1 += SpMM(vals, B0)
//   B0 = relu(B1 @ W1b^T + b1b)
//   B1 = B0 @ W2b^T + b2b          (h2)
//   B0 = relu(B1 @ Wf1^T + bf1)
//   out = B0 @ Wf2^T + bf2
// Workspace requirement: 2 * N * 256 * 4 bytes (~205 MB).

#define DIN  128
#define DH   256
#define DOUT 64

typedef __attribute__((ext_vector_type(2))) float v2f;
typedef __attribute__((ext_vector_type(8))) float v8f;

// dst = (1 + eps[0]) * src, float4-vectorized. Seeds the SpMM accumulator so the
// eps-residual is fused into the scatter-add for free.
__global__ __launch_bounds__(256) void k_scale_init(
    const float* __restrict__ src, const float* __restrict__ eps,
    float* __restrict__ dst, int n4)
{
    int i = blockIdx.x * 256 + threadIdx.x;
    if (i >= n4) return;
    const float s = 1.0f + eps[0];
    float4 v = reinterpret_cast<const float4*>(src)[i];
    v.x *= s; v.y *= s; v.z *= s; v.w *= s;
    reinterpret_cast<float4*>(dst)[i] = v;
}

// y[row[e], :] += vals[e] * x[col[e], :]
// Edge-parallel: D/4 lanes per edge, float4 gather (coalesced 512B/1KB per edge),
// 4x global_atomic_add_f32 scatter. Node matrices fit in 192MB L2 -> L2-resident.
template<int D>
__global__ __launch_bounds__(256) void k_spmm_atomic(
    const int* __restrict__ row, const int* __restrict__ col,
    const float* __restrict__ vals, const float* __restrict__ x,
    float* __restrict__ y, int E)
{
    constexpr int CH = D / 4;
    int tid = blockIdx.x * 256 + threadIdx.x;
    int e = tid / CH;
    int c = tid - e * CH;
    if (e >= E) return;
    const int   cs = col[e];
    const int   rs = row[e];
    const float v  = vals[e];
    float4 xv = reinterpret_cast<const float4*>(x + (size_t)cs * D)[c];
    float* dst = y + (size_t)rs * D + c * 4;
    unsafeAtomicAdd(dst + 0, v * xv.x);
    unsafeAtomicAdd(dst + 1, v * xv.y);
    unsafeAtomicAdd(dst + 2, v * xv.z);
    unsafeAtomicAdd(dst + 3, v * xv.w);
}

// C[N,FOUT] = act(A[N,FIN] @ W[FOUT,FIN]^T + bias), full fp32 via
// v_wmma_f32_16x16x4_f32. One wave owns a 16x64 tile (4 accumulators),
// K-loop step 4. A-fragment (16x4 MxK): lanes0-15 get K={k,k+1}, lanes16-31
// K={k+2,k+3} -> one float2 load per lane fills both A VGPRs; B[k][o]=W[o][k]
// uses the identical addressing pattern on W.
template<int FIN, int FOUT, bool RELU>
__global__ __launch_bounds__(256) void k_gemm_wmma(
    const float* __restrict__ A, const float* __restrict__ W,
    const float* __restrict__ bias, float* __restrict__ C, int nRows)
{
    constexpr int CG = FOUT / 64;                 // 64-col groups per row tile
    const int wave = threadIdx.x >> 5;
    const int lane = threadIdx.x & 31;
    const int tile = blockIdx.x * 8 + wave;
    const int totalTiles = (nRows >> 4) * CG;     // nRows divisible by 16
    if (tile >= totalTiles) return;               // uniform per wave: EXEC stays all-1s
    const int rowTile = tile / CG;
    const int cg      = tile - rowTile * CG;
    const int row0 = rowTile << 4;
    const int col0 = cg << 6;

    const int m    = lane & 15;                   // M (for A) / N-col (for B,C)
    const int kpar = (lane >> 4) << 1;            // half-wave K offset: 0 or 2

    const float* aPtr = A + (size_t)(row0 + m) * FIN + kpar;
    const float* wPtr = W + (size_t)(col0 + m) * FIN + kpar;

    v8f acc0 = {}, acc1 = {}, acc2 = {}, acc3 = {};
#pragma unroll 4
    for (int k = 0; k < FIN; k += 4) {
        v2f a  = *(const v2f*)(aPtr + k);
        v2f b0 = *(const v2f*)(wPtr + k);
        v2f b1 = *(const v2f*)(wPtr + 16 * FIN + k);
        v2f b2 = *(const v2f*)(wPtr + 32 * FIN + k);
        v2f b3 = *(const v2f*)(wPtr + 48 * FIN + k);
        acc0 = __builtin_amdgcn_wmma_f32_16x16x4_f32(false, a, false, b0, (short)0, acc0, false, false);
        acc1 = __builtin_amdgcn_wmma_f32_16x16x4_f32(false, a, false, b1, (short)0, acc1, false, false);
        acc2 = __builtin_amdgcn_wmma_f32_16x16x4_f32(false, a, false, b2, (short)0, acc2, false, false);
        acc3 = __builtin_amdgcn_wmma_f32_16x16x4_f32(false, a, false, b3, (short)0, acc3, false, false);
    }

    // C/D layout: VGPR r -> M = r (lanes 0-15) or r+8 (lanes 16-31), N = lane%16.
    const int mOff = (lane >> 4) << 3;            // 0 or 8
    const float bv0 = bias[col0 + m];
    const float bv1 = bias[col0 + 16 + m];
    const float bv2 = bias[col0 + 32 + m];
    const float bv3 = bias[col0 + 48 + m];
#pragma unroll
    for (int r = 0; r < 8; ++r) {
        float* cp = C + (size_t)(row0 + mOff + r) * FOUT + col0 + m;
        float o0 = acc0[r] + bv0;
        float o1 = acc1[r] + bv1;
        float o2 = acc2[r] + bv2;
        float o3 = acc3[r] + bv3;
        if (RELU) {
            o0 = fmaxf(o0, 0.0f); o1 = fmaxf(o1, 0.0f);
            o2 = fmaxf(o2, 0.0f); o3 = fmaxf(o3, 0.0f);
        }
        cp[0]  = o0;
        cp[16] = o1;
        cp[32] = o2;
        cp[48] = o3;
    }
}

extern "C" void kernel_launch(void* const* d_in, const int* in_sizes, int n_in,
                              void* d_out, int out_size, void* d_ws, size_t ws_size,
                              hipStream_t stream)
{
    (void)n_in; (void)out_size; (void)ws_size;

    const float* x    = (const float*)d_in[0];
    const int*   row  = (const int*)  d_in[1];
    const int*   col  = (const int*)  d_in[2];
    const float* vals = (const float*)d_in[3];
    const float* eps0 = (const float*)d_in[4];
    const float* W1a  = (const float*)d_in[5];
    const float* b1a  = (const float*)d_in[6];
    const float* W2a  = (const float*)d_in[7];
    const float* b2a  = (const float*)d_in[8];
    const float* eps1 = (const float*)d_in[9];
    const float* W1b  = (const float*)d_in[10];
    const float* b1b  = (const float*)d_in[11];
    const float* W2b  = (const float*)d_in[12];
    const float* b2b  = (const float*)d_in[13];
    const float* Wf1  = (const float*)d_in[14];
    const float* bf1  = (const float*)d_in[15];
    const float* Wf2  = (const float*)d_in[16];
    const float* bf2  = (const float*)d_in[17];

    const int N = in_sizes[0] / DIN;   // 100000
    const int E = in_sizes[1];         // 1600000

    float* B0  = (float*)d_ws;
    float* B1  = B0 + (size_t)N * DH;
    float* out = (float*)d_out;

    const int gemmTilesH = (N / 16) * (DH / 64);
    const int gemmBlksH  = (gemmTilesH + 7) / 8;
    const int gemmTilesO = (N / 16) * (DOUT / 64);
    const int gemmBlksO  = (gemmTilesO + 7) / 8;

    // ---- GIN layer 1 (d=128 -> 256 -> 256) ----
    {
        int n4 = N * DIN / 4;
        k_scale_init<<<(n4 + 255) / 256, 256, 0, stream>>>(x, eps0, B0, n4);
        int tot = E * (DIN / 4);
        k_spmm_atomic<DIN><<<(tot + 255) / 256, 256, 0, stream>>>(row, col, vals, x, B0, E);
        k_gemm_wmma<DIN, DH, true ><<<gemmBlksH, 256, 0, stream>>>(B0, W1a, b1a, B1, N);
        k_gemm_wmma<DH,  DH, false><<<gemmBlksH, 256, 0, stream>>>(B1, W2a, b2a, B0, N); // h1 -> B0
    }
    // ---- GIN layer 2 (d=256) ----
    {
        int n4 = N * DH / 4;
        k_scale_init<<<(n4 + 255) / 256, 256, 0, stream>>>(B0, eps1, B1, n4);
        int tot = E * (DH / 4);
        k_spmm_atomic<DH><<<(tot + 255) / 256, 256, 0, stream>>>(row, col, vals, B0, B1, E);
        k_gemm_wmma<DH, DH, true ><<<gemmBlksH, 256, 0, stream>>>(B1, W1b, b1b, B0, N);
        k_gemm_wmma<DH, DH, false><<<gemmBlksH, 256, 0, stream>>>(B0, W2b, b2b, B1, N);  // h2 -> B1
    }
    // ---- Head ----
    k_gemm_wmma<DH, DH,   true ><<<gemmBlksH, 256, 0, stream>>>(B1, Wf1, bf1, B0, N);
    k_gemm_wmma<DH, DOUT, false><<<gemmBlksO, 256, 0, stream>>>(B0, Wf2, bf2, out, N);
}